// wFMLayer_89670327205972
// MI455X (gfx1250) — compile-verified
//
#include <hip/hip_runtime.h>

typedef __attribute__((ext_vector_type(2))) float v2f;
typedef __attribute__((ext_vector_type(8))) float v8f;

#define B_    4
#define N_    4096
#define D_    25
#define C_    32
#define K_    16
#define OUT_  64
#define EPSF  1e-12f

#define ROWS  36   // LDS row stride in floats (32 + 4 pad, keeps 16B align, spreads banks)

// ---------------------------------------------------------------------------
// Kernel 0: normalize weights once into workspace.
//  ws1t : [K_][C_]  = (w1 / rownorm(w1))^T   (transposed so stage-1 reads are
//                     one contiguous float4 per (k, channel-quad))
//  wsw2 : [C_][OUT_] = w2 / colnorm(w2)
// ---------------------------------------------------------------------------
__global__ void wfm_normalize_kernel(const float* __restrict__ w1,
                                     const float* __restrict__ w2,
                                     float* __restrict__ ws1t,
                                     float* __restrict__ wsw2) {
  const int t = threadIdx.x;           // 64 threads, 1 block
  if (t < C_) {
    float s = 0.f;
#pragma unroll
    for (int k = 0; k < K_; ++k) { float v = w1[t * K_ + k]; s += v * v; }
    const float inv = 1.0f / fmaxf(sqrtf(s), EPSF);
#pragma unroll
    for (int k = 0; k < K_; ++k) ws1t[k * C_ + t] = w1[t * K_ + k] * inv;
  }
  {
    float s = 0.f;
#pragma unroll
    for (int c = 0; c < C_; ++c) { float v = w2[c * OUT_ + t]; s += v * v; }
    const float inv = 1.0f / fmaxf(sqrtf(s), EPSF);
#pragma unroll
    for (int c = 0; c < C_; ++c) wsw2[c * OUT_ + t] = w2[c * OUT_ + t] * inv;
  }
}

// ---------------------------------------------------------------------------
// Main kernel: one wave32 per (b, n) point; 4 waves per block.
// Stage 1: gather + K-contraction -> weighted[25][32] (registers -> LDS)
// Stage 2: weighted(25x32) @ w2n(32x64) via v_wmma_f32_16x16x4_f32
// ---------------------------------------------------------------------------
__global__ __launch_bounds__(128)
void wfm_main_kernel(const float* __restrict__ x,
                     const long long* __restrict__ knn,   // int64 indices
                     const float* __restrict__ ws1t,
                     const float* __restrict__ wsw2,
                     float* __restrict__ out) {
  __shared__ float Wlds[4][32 * ROWS];

  const int lane = threadIdx.x & 31;
  const int wv   = threadIdx.x >> 5;
  const int p    = blockIdx.x * 4 + wv;   // 0 .. B_*N_-1 (grid sized exactly)
  const int b    = p >> 12;               // / N_
  // n = p & 4095 is implicit in the flat index p

  float* Wl = &Wlds[wv][0];

  // Zero pad rows 25..31 (read by the mt=1 A-tiles).
  for (int i = lane; i < 7 * ROWS; i += 32) Wl[25 * ROWS + i] = 0.f;

  // ---- Stage 1: lane = (row-offset r, channel-quad q) --------------------
  const int q = lane & 7;    // channel quad, c0 = 4*q
  const int r = lane >> 3;   // d offset 0..3; this lane owns d = r, r+4, ...

  float4 acc[7];
#pragma unroll
  for (int j = 0; j < 7; ++j) acc[j] = make_float4(0.f, 0.f, 0.f, 0.f);

  const long long* kbase = knn + (size_t)p * K_;
  for (int k = 0; k < K_; ++k) {
    const long long m   = kbase[k];
    const float*    xb  = x + ((size_t)b * N_ + (size_t)m) * (D_ * C_) + q * 4;
    const float4    wq  = *reinterpret_cast<const float4*>(ws1t + k * C_ + q * 4);
#pragma unroll
    for (int j = 0; j < 7; ++j) {
      const int d = r + 4 * j;
      if (d < D_) {
        const float4 xv = *reinterpret_cast<const float4*>(xb + d * C_);
        acc[j].x = fmaf(xv.x, wq.x, acc[j].x);
        acc[j].y = fmaf(xv.y, wq.y, acc[j].y);
        acc[j].z = fmaf(xv.z, wq.z, acc[j].z);
        acc[j].w = fmaf(xv.w, wq.w, acc[j].w);
      }
    }
  }

#pragma unroll
  for (int j = 0; j < 7; ++j) {
    const int d = r + 4 * j;
    if (d < D_)
      *reinterpret_cast<float4*>(Wl + d * ROWS + q * 4) = acc[j];
  }

  __syncthreads();   // cross-lane LDS reshuffle into WMMA fragment layout

  // ---- Stage 2: WMMA f32 16x16x4 ----------------------------------------
  // A 16x4 f32 layout: lanes 0-15 hold {K=0,K=1}, lanes 16-31 hold {K=2,K=3}.
  // C/D layout: VGPR v -> row v (lanes 0-15) / row v+8 (lanes 16-31).
  const int on    = lane & 15;       // output column within 16-tile
  const int chalf = lane >> 4;       // 0 or 1: K-subpair selector

  // Preload all B fragments (w2n is tiny, L2-resident) — reused by both mt.
  v2f bfrag[8][4];
#pragma unroll
  for (int kc = 0; kc < 8; ++kc) {
    const int crow = kc * 4 + chalf * 2;
#pragma unroll
    for (int nt = 0; nt < 4; ++nt) {
      const int o = nt * 16 + on;
      v2f bv;
      bv[0] = wsw2[crow * OUT_ + o];
      bv[1] = wsw2[(crow + 1) * OUT_ + o];
      bfrag[kc][nt] = bv;
    }
  }

#pragma unroll
  for (int mt = 0; mt < 2; ++mt) {
    const int arow = mt * 16 + (lane & 15);

    v2f afrag[8];
#pragma unroll
    for (int kc = 0; kc < 8; ++kc) {
      v2f av;
      av[0] = Wl[arow * ROWS + kc * 4 + chalf * 2 + 0];
      av[1] = Wl[arow * ROWS + kc * 4 + chalf * 2 + 1];
      afrag[kc] = av;
    }

#pragma unroll
    for (int nt = 0; nt < 4; ++nt) {
      v8f c8 = {};
#pragma unroll
      for (int kc = 0; kc < 8; ++kc) {
        c8 = __builtin_amdgcn_wmma_f32_16x16x4_f32(
            /*neg_a=*/false, afrag[kc],
            /*neg_b=*/false, bfrag[kc][nt],
            /*c_mod=*/(short)0, c8,
            /*reuse_a=*/false, /*reuse_b=*/false);
      }
#pragma unroll
      for (int v = 0; v < 8; ++v) {
        const int drow = mt * 16 + chalf * 8 + v;
        if (drow < D_)
          out[((size_t)p * D_ + drow) * OUT_ + nt * 16 + on] = c8[v];
      }
    }
  }
}

// ---------------------------------------------------------------------------
extern "C" void kernel_launch(void* const* d_in, const int* in_sizes, int n_in,
                              void* d_out, int out_size, void* d_ws, size_t ws_size,
                              hipStream_t stream) {
  const float*     x   = (const float*)d_in[0];
  const long long* knn = (const long long*)d_in[1];   // int64 per reference
  const float*     w1  = (const float*)d_in[2];
  const float*     w2  = (const float*)d_in[3];
  float*           out = (float*)d_out;

  float* ws1t = (float*)d_ws;          // K_*C_   = 512 floats
  float* wsw2 = ws1t + K_ * C_;        // C_*OUT_ = 2048 floats (10 KB total)

  wfm_normalize_kernel<<<1, 64, 0, stream>>>(w1, w2, ws1t, wsw2);
  wfm_main_kernel<<<(B_ * N_) / 4, 128, 0, stream>>>(x, knn, ws1t, wsw2, out);
}